// SingleDeformableLayer_31112743092640
// MI455X (gfx1250) — compile-verified
//
#include <hip/hip_runtime.h>
#include <hip/hip_bf16.h>
#include <math.h>

// ---------------------------------------------------------------------------
// Deformable conv forward for MI455X (gfx1250, wave32).
//   B=4, C=64, H=W=128, CO=64, K=3, PAD=1, STRIDE=1  ->  Ho=Wo=128
// Stage 1: offset conv (plain VALU, L2-bound, ~1.4 GFLOP)
// Stage 2: bilinear-gather im2col tile into LDS + fp32 WMMA GEMM
//          Out(64 x 65536) = Wdef(64 x 576) x S(576 x 65536)
//          using V_WMMA_F32_16X16X4_F32 (fp32 matches reference precision).
// ---------------------------------------------------------------------------

typedef __attribute__((ext_vector_type(2))) float v2f;
typedef __attribute__((ext_vector_type(8))) float v8f;

#define BATCH 4
#define CIN   64
#define HDIM  128
#define WDIM  128
#define HW    (HDIM * WDIM)       // 16384
#define NOFF  18                  // 2*K*K offset channels
#define COUT  64
#define KKT   9                   // K*K taps
#define KDIM  (CIN * KKT)         // 576 (flattened as c*9+kk, matching reshape(Co,C,KK))
#define KPAIRS (KDIM / 2)         // 288
#define TP    16                  // output positions per workgroup (N tile)
#define S2STRIDE 34               // floats per K-pair row: 16 pos * 2 + 2 pad (bank skew)

// ---------------------------------------------------------------------------
// Kernel 1: offset conv.  off[b, 2kk(+1), h, w] = conv3x3(x, w_off) + b_off
// One thread per (b,h,w); all 18 output channels accumulated in registers.
// w_off (18*64*9 = 10368 floats = 40.5 KB) staged in LDS, broadcast-read.
// ---------------------------------------------------------------------------
__global__ __launch_bounds__(256) void offset_conv_kernel(
    const float* __restrict__ x, const float* __restrict__ w_off,
    const float* __restrict__ b_off, float* __restrict__ off) {
  __shared__ float wsm[NOFF * CIN * KKT];
  const int t = threadIdx.x;
  for (int i = t; i < NOFF * CIN * KKT; i += 256) wsm[i] = w_off[i];
  __syncthreads();

  const int p = blockIdx.x * 256 + t;       // 0 .. 65535
  const int b = p >> 14;
  const int hw = p & (HW - 1);
  const int h = hw >> 7;
  const int w = hw & (WDIM - 1);

  float acc[NOFF];
#pragma unroll
  for (int o = 0; o < NOFF; ++o) acc[o] = b_off[o];

  const float* xb = x + b * CIN * HW;
  for (int c = 0; c < CIN; ++c) {
    const float* xc = xb + c * HW;
#pragma unroll
    for (int ky = 0; ky < 3; ++ky) {
      const int iy = h + ky - 1;
      if (iy < 0 || iy >= HDIM) continue;
#pragma unroll
      for (int kx = 0; kx < 3; ++kx) {
        const int ix = w + kx - 1;
        if (ix < 0 || ix >= WDIM) continue;
        const float xv = xc[iy * WDIM + ix];
        const int widx = c * KKT + ky * 3 + kx;
#pragma unroll
        for (int o = 0; o < NOFF; ++o)
          acc[o] = fmaf(xv, wsm[o * KDIM + widx], acc[o]);
      }
    }
  }
  float* op = off + b * NOFF * HW + hw;
#pragma unroll
  for (int o = 0; o < NOFF; ++o) op[o * HW] = acc[o];
}

// ---------------------------------------------------------------------------
// Kernel 2: bilinear gather + fp32 WMMA GEMM.
// Block = 128 threads (4 waves), one block per TP=16 output positions.
//  Phase 1a: per (tap,pos) bilinear corner indices/weights -> LDS (chan-invariant)
//  Phase 1b: build S tile [576 x 16] in LDS (K-pair interleaved for 8B reads)
//  Phase 2 : wave wv owns D tile co[wv*16 .. +15] x pos[0..15];
//            144 steps of v_wmma_f32_16x16x4_f32 accumulating over K=576.
// ---------------------------------------------------------------------------
__global__ __launch_bounds__(128) void deform_wmma_kernel(
    const float* __restrict__ x, const float* __restrict__ off,
    const float* __restrict__ w_def, float* __restrict__ out) {
  __shared__ float S2[KPAIRS * S2STRIDE];   // 39168 B
  __shared__ int   gidx[KKT * TP * 4];      //  2304 B  clamped plane indices
  __shared__ float gwt [KKT * TP * 4];      //  2304 B  bilinear weights (0 if OOB)
  __shared__ int   bbase[TP];               // b*CIN*HW per position

  const int t = threadIdx.x;
  const int p0 = blockIdx.x * TP;

  // ---- Phase 1a: sampling geometry (144 (tap,pos) pairs) -------------------
  for (int i = t; i < KKT * TP; i += 128) {
    const int kk = i / TP, tp = i - kk * TP;
    const int p = p0 + tp;
    const int b = p >> 14;
    const int hw = p & (HW - 1);
    const int h = hw >> 7;
    const int w = hw & (WDIM - 1);
    if (kk == 0) bbase[tp] = b * CIN * HW;

    const int ky = kk / 3, kx = kk - ky * 3;
    const float* ob = off + b * NOFF * HW + hw;
    const float py = ob[(2 * kk)     * HW] + (float)(ky + h - 1);  // STRIDE=1,PAD=1
    const float px = ob[(2 * kk + 1) * HW] + (float)(kx + w - 1);
    const float y0f = floorf(py), x0f = floorf(px);
    const float fy = py - y0f, fx = px - x0f;
    const int y0 = (int)y0f, x0 = (int)x0f;

    const int base4 = i * 4;
#pragma unroll
    for (int j = 0; j < 4; ++j) {
      const int iy = y0 + (j >> 1);
      const int ix = x0 + (j & 1);
      const float wy = (j >> 1) ? fy : (1.0f - fy);
      const float wx = (j & 1)  ? fx : (1.0f - fx);
      const bool valid = (iy >= 0) & (iy < HDIM) & (ix >= 0) & (ix < WDIM);
      const int ciy = min(max(iy, 0), HDIM - 1);
      const int cix = min(max(ix, 0), WDIM - 1);
      gidx[base4 + j] = ciy * WDIM + cix;
      gwt [base4 + j] = valid ? (wy * wx) : 0.0f;
    }
  }
  __syncthreads();

  // ---- Phase 1b: gather S tile [KDIM x TP] into LDS ------------------------
  for (int i = t; i < KDIM * TP; i += 128) {        // 72 iters/thread
    const int k  = i >> 4;          // 0..575  (k = c*9 + kk, matches w_def flatten)
    const int tp = i & (TP - 1);
    const int c  = k / KKT;
    const int kk = k - c * KKT;
    const float* xp = x + bbase[tp] + c * HW;
    const int b4 = (kk * TP + tp) * 4;
    const float s = gwt[b4 + 0] * xp[gidx[b4 + 0]]
                  + gwt[b4 + 1] * xp[gidx[b4 + 1]]
                  + gwt[b4 + 2] * xp[gidx[b4 + 2]]
                  + gwt[b4 + 3] * xp[gidx[b4 + 3]];
    S2[(k >> 1) * S2STRIDE + tp * 2 + (k & 1)] = s;   // K-pair interleave
  }
  __syncthreads();

  // ---- Phase 2: fp32 WMMA GEMM --------------------------------------------
  // A (16x4 f32): lanes 0-15 hold rows M (co), VGPR v -> K=v; lanes 16-31 -> K=2+v
  // B (4x16 f32): lanes 0-15 hold cols N (pos), VGPR v -> K=v; lanes 16-31 -> K=2+v
  const int wv = t >> 5;              // wave id 0..3 -> co tile
  const int lane = t & 31;
  const int lane16 = lane & 15;
  const int halfk = lane >> 4;        // selects K pair {0,1} vs {2,3}
  const int co_base = wv * 16;

  v8f acc = {};
  const float* wd = w_def + (co_base + lane16) * KDIM + halfk * 2;
  const float* sp = S2 + halfk * S2STRIDE + lane16 * 2;

#pragma unroll 4
  for (int kc = 0; kc < KDIM / 4; ++kc) {           // 144 WMMA steps
    const v2f a = *(const v2f*)(wd + kc * 4);                 // global_load_b64
    const v2f bm = *(const v2f*)(sp + kc * 2 * S2STRIDE);     // ds_load_b64
    acc = __builtin_amdgcn_wmma_f32_16x16x4_f32(
        /*neg_a=*/false, a, /*neg_b=*/false, bm,
        /*c_mod=*/(short)0, acc, /*reuse_a=*/false, /*reuse_b=*/false);
  }

  // ---- Writeback: D 16x16 f32 layout: VGPR r -> M=r (lanes 0-15), r+8 (16-31)
  const int p = p0 + lane16;          // N = pos = lane16
  const int b = p >> 14;
  const int hw = p & (HW - 1);
#pragma unroll
  for (int r = 0; r < 8; ++r) {
    const int co = co_base + r + (halfk ? 8 : 0);
    out[(b * COUT + co) * HW + hw] = acc[r];
  }
}

// ---------------------------------------------------------------------------
extern "C" void kernel_launch(void* const* d_in, const int* in_sizes, int n_in,
                              void* d_out, int out_size, void* d_ws, size_t ws_size,
                              hipStream_t stream) {
  const float* x     = (const float*)d_in[0];   // [4,64,128,128]
  const float* w_off = (const float*)d_in[1];   // [18,64,3,3]
  const float* b_off = (const float*)d_in[2];   // [18]
  const float* w_def = (const float*)d_in[3];   // [64,64,3,3] -> (64,576)
  float* out = (float*)d_out;                   // [4,64,128,128]
  float* off = (float*)d_ws;                    // [4,18,128,128] scratch (4.7 MB)

  (void)in_sizes; (void)n_in; (void)out_size; (void)ws_size;

  // Stage 1: offset conv -> d_ws
  offset_conv_kernel<<<(BATCH * HW) / 256, 256, 0, stream>>>(x, w_off, b_off, off);

  // Stage 2: deformable gather + WMMA GEMM -> d_out
  deform_wmma_kernel<<<(BATCH * HW) / TP, 128, 0, stream>>>(x, off, w_def, out);
}